// TensorSSM_82849919140189
// MI455X (gfx1250) — compile-verified
//
#include <hip/hip_runtime.h>

#define T_LEN 1024
#define D 256
#define DD (D * D)

typedef __attribute__((ext_vector_type(2))) float v2f;
typedef __attribute__((ext_vector_type(8))) float v8f;

// ---------------------------------------------------------------------------
// Kernel 1: inclusive prefix sum over time for one chunk of timesteps.
// S_t = carry + sum_{s in chunk, s<=t} xs_s ; writes chunk-local S into Sc,
// updates carry at the end. One thread owns 4 consecutive elements (float4).
// Grid: 64 blocks x 256 threads = 16384 threads x 4 floats = 65536 elements.
// ---------------------------------------------------------------------------
__global__ __launch_bounds__(256) void tssm_prefix(const float* __restrict__ xs,
                                                   float* __restrict__ carry,
                                                   float* __restrict__ Sc,
                                                   int t0, int tn, int first) {
  const int idx4 = (blockIdx.x * 256 + threadIdx.x) * 4;
  float4 acc;
  if (first) {
    acc = make_float4(0.f, 0.f, 0.f, 0.f);
  } else {
    acc = *(const float4*)(carry + idx4);
  }
  const float* xp = xs + (size_t)t0 * DD + idx4;
  float* sp = Sc + idx4;
  for (int t = 0; t < tn; ++t) {
    float4 x = *(const float4*)(xp + (size_t)t * DD);
    acc.x += x.x; acc.y += x.y; acc.z += x.z; acc.w += x.w;
    *(float4*)(sp + (size_t)t * DD) = acc;
  }
  *(float4*)(carry + idx4) = acc;
}

// ---------------------------------------------------------------------------
// Kernel 2: out_t = S_t @ S_t (256x256x256 fp32 GEMM per timestep).
// Block = 256 threads (8 waves), computes a 128-row x 256-col half of out_t.
// LDS: A panel 128x32 (pad 36 -> conflict-free b64 A-frag loads),
//      B panel  32x256 (pad 264 -> half-waves hit disjoint bank sets).
// Each wave: 4x4 grid of 16x16 WMMA tiles (64x64), K in steps of 4 via
// v_wmma_f32_16x16x4_f32 -> 16 WMMAs per ~8 DS ops per K-step.
// ---------------------------------------------------------------------------
__global__ __launch_bounds__(256) void tssm_mm(const float* __restrict__ S,
                                               float* __restrict__ out) {
  __shared__ float As[128][36];    // 18432 B, row stride 144 B (16B aligned)
  __shared__ float Bs[32][264];    // 33792 B, row stride 1056 B (16B aligned)

  const int t = blockIdx.y;
  const float* St = S + (size_t)t * DD;
  float* Ot = out + (size_t)t * DD;

  const int brow = blockIdx.x * 128;   // row half of out_t; cols = all 256

  const int tid  = threadIdx.x;
  const int w    = tid >> 5;
  const int lane = tid & 31;
  const int lm   = lane & 15;      // row/col within 16x16 tile
  const int lh   = lane >> 4;      // half-wave selects K pair / M+8
  const int wrB  = (w >> 2) * 64;  // wave tile-row base within 128
  const int wcB  = (w & 3) * 64;   // wave tile-col base within 256

  v8f acc[4][4];
  {
    v8f z = {0.f, 0.f, 0.f, 0.f, 0.f, 0.f, 0.f, 0.f};
#pragma unroll
    for (int m = 0; m < 4; ++m)
#pragma unroll
      for (int n = 0; n < 4; ++n) acc[m][n] = z;
  }

  for (int kp = 0; kp < D; kp += 32) {
    // ---- stage A (rows brow..+127, cols kp..+31): 4 float4 / thread ----
#pragma unroll
    for (int i = 0; i < 4; ++i) {
      int e  = tid + i * 256;                // 0..1023
      int r  = e >> 3;                       // 0..127
      int kq = (e & 7) << 2;                 // 0..28
      float4 va = *(const float4*)(St + (size_t)(brow + r) * D + kp + kq);
      *(float4*)&As[r][kq] = va;
    }
    // ---- stage B (rows kp..+31, cols 0..255): 8 float4 / thread ----
#pragma unroll
    for (int i = 0; i < 8; ++i) {
      int e  = tid + i * 256;                // 0..2047
      int kk = e >> 6;                       // 0..31
      int c4 = (e & 63) << 2;                // 0..252
      float4 vb = *(const float4*)(St + (size_t)(kp + kk) * D + c4);
      *(float4*)&Bs[kk][c4] = vb;
    }
    __syncthreads();

    // ---- 8 K-steps of 4, 16 WMMAs each ----
#pragma unroll
    for (int kk = 0; kk < 32; kk += 4) {
      const int k0 = kk + 2 * lh;
      v2f a[4], b[4];
#pragma unroll
      for (int m = 0; m < 4; ++m) {
        int row = wrB + m * 16 + lm;
        a[m].x = As[row][k0];
        a[m].y = As[row][k0 + 1];
      }
#pragma unroll
      for (int n = 0; n < 4; ++n) {
        int col = wcB + n * 16 + lm;
        b[n].x = Bs[k0][col];
        b[n].y = Bs[k0 + 1][col];
      }
#pragma unroll
      for (int m = 0; m < 4; ++m)
#pragma unroll
        for (int n = 0; n < 4; ++n)
          acc[m][n] = __builtin_amdgcn_wmma_f32_16x16x4_f32(
              false, a[m], false, b[n], (short)0, acc[m][n], false, false);
    }
    __syncthreads();
  }

  // ---- store: D VGPR r holds row (r + 8*lh), col lm of each 16x16 tile ----
#pragma unroll
  for (int m = 0; m < 4; ++m)
#pragma unroll
    for (int n = 0; n < 4; ++n) {
      int col = wcB + n * 16 + lm;
#pragma unroll
      for (int r = 0; r < 8; ++r) {
        int row = brow + wrB + m * 16 + r + 8 * lh;
        Ot[(size_t)row * D + col] = acc[m][n][r];
      }
    }
}

// ---------------------------------------------------------------------------
// Host: out_t = (cumsum_{s<=t} xs)^2 by telescoping; chunk time axis so that
// workspace holds carry (256KB) + chunk * 256KB of S.
// ---------------------------------------------------------------------------
extern "C" void kernel_launch(void* const* d_in, const int* in_sizes, int n_in,
                              void* d_out, int out_size, void* d_ws, size_t ws_size,
                              hipStream_t stream) {
  const float* xs = (const float*)d_in[0];
  float* out = (float*)d_out;
  float* carry = (float*)d_ws;
  float* Sc = carry + DD;

  const size_t matBytes = (size_t)DD * sizeof(float);
  size_t mats = ws_size / matBytes;
  size_t cap = (mats > 1) ? (mats - 1) : 1;
  if (cap > (size_t)T_LEN) cap = T_LEN;
  const int chunk = (int)cap;

  for (int t0 = 0; t0 < T_LEN; t0 += chunk) {
    const int tn = (T_LEN - t0 < chunk) ? (T_LEN - t0) : chunk;
    tssm_prefix<<<64, 256, 0, stream>>>(xs, carry, Sc, t0, tn, t0 == 0 ? 1 : 0);
    dim3 grid(2, tn);
    tssm_mm<<<grid, 256, 0, stream>>>(Sc, out + (size_t)t0 * DD);
  }
}